// DBNSigma_17987323036450
// MI455X (gfx1250) — compile-verified
//
#include <hip/hip_runtime.h>

// Decorrelated BatchNorm (group ZCA whitening), G=16 groups x CG=16 channels,
// X = [64, 256, 56, 56] f32.  Bandwidth-bound (~616 MB min traffic -> ~26us
// at 23.3 TB/s).  Covariance and whitening GEMMs mapped onto
// V_WMMA_F32_16X16X4_F32 (fp32 16x16 tile, K=4).

typedef __attribute__((ext_vector_type(2))) float v2f;
typedef __attribute__((ext_vector_type(8))) float v8f;

#define GRP   16          // groups
#define CG    16          // channels per group
#define CCH   256         // total channels
#define NN    64          // batch
#define HW    3136        // 56*56
#define MTOT  200704      // NN*HW samples per channel
#define BPG   64          // blocks per group (one N-slice each)
#define EPSV  1e-3f

// workspace float offsets
#define WS_PSIG 0                            // G*BPG*256 partial raw 2nd moments
#define WS_PSUM (GRP*BPG*256)                // G*BPG*16 partial channel sums
#define WS_WM   (WS_PSUM + GRP*BPG*16)       // G*256 whitening matrices (weight folded)
#define WS_SH   (WS_WM + GRP*256)            // G*16 shifts (bias - W*mean)

// ---------------------------------------------------------------------------
// Pass 1: per (group, N-slice) partial sums + raw 16x16 second moments.
// A and B operands of the WMMA are the SAME register image (symmetric xx^T):
//   lane c (<16):  x[c][m0+0], x[c][m0+1]   (v2f contiguous load)
//   lane 16+c:     x[c][m0+2], x[c][m0+3]
// matching the 32-bit 16x4 A layout and 4x16 B layout.
// ---------------------------------------------------------------------------
__global__ __launch_bounds__(256) void dbn_pass1(const float* __restrict__ X,
                                                 float* __restrict__ ws)
{
    __shared__ float s_sig[8 * 256];
    __shared__ float s_sum[8 * 32];

    const int g    = blockIdx.x / BPG;
    const int blk  = blockIdx.x % BPG;       // N-slice index
    const int tid  = threadIdx.x;
    const int lane = tid & 31;
    const int wave = tid >> 5;
    const int c    = lane & 15;
    const int hi   = lane >> 4;              // 0: K=0..1, 1: K=2..3

    const float* base = X + ((size_t)(blk * CCH + g * CG + c)) * HW;
    const int mstart  = wave * (HW / 8) + (hi << 1);   // 392 m per wave

    v8f  acc  = {};
    float lsum = 0.f;
    for (int i = 0; i < 98; ++i) {           // 98 * 4 = 392 m-positions
        const v2f a = *(const v2f*)(base + mstart + i * 4);
        lsum += a.x + a.y;
        acc = __builtin_amdgcn_wmma_f32_16x16x4_f32(
            false, a, false, a, (short)0, acc, false, false);
    }

    // dump v8f accumulator (WMMA C layout) to LDS row-major
#pragma unroll
    for (int r = 0; r < 8; ++r) {
        const int row = r + (hi << 3);
        s_sig[wave * 256 + row * 16 + c] = acc[r];
    }
    s_sum[wave * 32 + lane] = lsum;
    __syncthreads();

    float tot = 0.f;
#pragma unroll
    for (int w = 0; w < 8; ++w) tot += s_sig[w * 256 + tid];
    ws[WS_PSIG + ((size_t)(g * BPG + blk)) * 256 + tid] = tot;

    if (tid < 16) {
        float s = 0.f;
#pragma unroll
        for (int w = 0; w < 8; ++w)
            s += s_sum[w * 32 + tid] + s_sum[w * 32 + tid + 16];
        ws[WS_PSUM + (g * BPG + blk) * 16 + tid] = s;
    }
}

// ---------------------------------------------------------------------------
// Pass 2: one block per group.  Deterministic reduction of partials,
// Sigma = E[xx^T] - mu mu^T + eps I, then Newton-Schulz iteration for
// Sigma^{-1/2} (trace-normalized, SPD-convergent).  Folds `weight` into the
// whitening matrix rows and precomputes shift = bias - W*mu.
// ---------------------------------------------------------------------------
__global__ __launch_bounds__(256) void dbn_pass2(float* __restrict__ ws,
                                                 const float* __restrict__ wgt,
                                                 const float* __restrict__ bia)
{
    __shared__ float Am[256], Ym[256], Zm[256], Tm[256], U1[256], U2[256];
    __shared__ float meanS[16];
    __shared__ float trS;

    const int g = blockIdx.x;
    const int t = threadIdx.x;
    const int c = t >> 4, d = t & 15;
    const float invm = 1.0f / (float)MTOT;

    if (t < 16) {
        float s = 0.f;
        for (int b = 0; b < BPG; ++b) s += ws[WS_PSUM + (g * BPG + b) * 16 + t];
        meanS[t] = s * invm;
    }
    float raw = 0.f;
    for (int b = 0; b < BPG; ++b)
        raw += ws[WS_PSIG + ((size_t)(g * BPG + b)) * 256 + t];
    __syncthreads();

    Am[t] = raw * invm - meanS[c] * meanS[d] + ((c == d) ? EPSV : 0.f);
    __syncthreads();

    if (t == 0) {
        float tr = 0.f;
        for (int i = 0; i < 16; ++i) tr += Am[i * 17];
        trS = tr;
    }
    __syncthreads();

    const float invtr = 1.0f / trS;
    Ym[t] = Am[t] * invtr;
    Zm[t] = (c == d) ? 1.f : 0.f;
    __syncthreads();

    for (int it = 0; it < 12; ++it) {
        float zy = 0.f;
#pragma unroll
        for (int k = 0; k < 16; ++k) zy += Zm[c * 16 + k] * Ym[k * 16 + d];
        Tm[t] = ((c == d) ? 1.5f : 0.f) - 0.5f * zy;
        __syncthreads();
        float yt = 0.f, tz = 0.f;
#pragma unroll
        for (int k = 0; k < 16; ++k) {
            yt += Ym[c * 16 + k] * Tm[k * 16 + d];
            tz += Tm[c * 16 + k] * Zm[k * 16 + d];
        }
        U1[t] = yt;
        U2[t] = tz;
        __syncthreads();
        Ym[t] = U1[t];
        Zm[t] = U2[t];
        __syncthreads();
    }

    // W = weight[c] * Z / sqrt(tr)   (A^{-1/2} = Z / sqrt(tr))
    const float wmv = Zm[t] * rsqrtf(trS) * wgt[g * CG + c];
    ws[WS_WM + g * 256 + t] = wmv;
    U1[t] = wmv;
    __syncthreads();
    if (t < 16) {
        float sh = bia[g * CG + t];
#pragma unroll
        for (int k = 0; k < 16; ++k) sh -= U1[t * 16 + k] * meanS[k];
        ws[WS_SH + g * 16 + t] = sh;
    }
}

// ---------------------------------------------------------------------------
// Pass 3: out[c][m] = sum_d W[c][d]*x[d][m] + shift[c], tiled as
// 16x16 output tiles = 4 chained wmma_f32_16x16x4_f32 (K = channel dim).
// W held in registers (A operand, 4 K-slices), x loaded 64B-contiguous per
// channel per tile.  16-aligned m-tiles never cross an N-slice (3136 % 16==0).
// ---------------------------------------------------------------------------
__global__ __launch_bounds__(256) void dbn_pass3(const float* __restrict__ X,
                                                 const float* __restrict__ ws,
                                                 float* __restrict__ out)
{
    const int g    = blockIdx.x / BPG;
    const int blk  = blockIdx.x % BPG;
    const int tid  = threadIdx.x;
    const int lane = tid & 31;
    const int wave = tid >> 5;
    const int lm   = lane & 15;
    const int hi   = lane >> 4;
    const int ko   = hi << 1;

    const float* wm = ws + WS_WM + g * 256;
    v2f a0, a1, a2, a3;
    a0.x = wm[lm * 16 + 0 + ko];  a0.y = wm[lm * 16 + 1 + ko];
    a1.x = wm[lm * 16 + 4 + ko];  a1.y = wm[lm * 16 + 5 + ko];
    a2.x = wm[lm * 16 + 8 + ko];  a2.y = wm[lm * 16 + 9 + ko];
    a3.x = wm[lm * 16 + 12 + ko]; a3.y = wm[lm * 16 + 13 + ko];

    float sh[8];
#pragma unroll
    for (int r = 0; r < 8; ++r) sh[r] = ws[WS_SH + g * 16 + r + (hi << 3)];

    const int wgid = blk * 8 + wave;         // 0..511 (wave-uniform)
    const int NT   = MTOT / 16;              // 12544 tiles per group

    for (int t = wgid; t < NT; t += BPG * 8) {
        const int n   = t / 196;             // 196 tiles per N-slice
        const int r16 = (t - n * 196) * 16;
        const float* xb = X + ((size_t)(n * CCH + g * CG)) * HW + r16 + lm;

        v8f acc = {};
        v2f b;
        b.x = xb[(0 + ko) * HW];  b.y = xb[(1 + ko) * HW];
        acc = __builtin_amdgcn_wmma_f32_16x16x4_f32(false, a0, false, b, (short)0, acc, false, false);
        b.x = xb[(4 + ko) * HW];  b.y = xb[(5 + ko) * HW];
        acc = __builtin_amdgcn_wmma_f32_16x16x4_f32(false, a1, false, b, (short)0, acc, false, false);
        b.x = xb[(8 + ko) * HW];  b.y = xb[(9 + ko) * HW];
        acc = __builtin_amdgcn_wmma_f32_16x16x4_f32(false, a2, false, b, (short)0, acc, false, false);
        b.x = xb[(12 + ko) * HW]; b.y = xb[(13 + ko) * HW];
        acc = __builtin_amdgcn_wmma_f32_16x16x4_f32(false, a3, false, b, (short)0, acc, false, false);

        float* ob = out + ((size_t)(n * CCH + g * CG)) * HW + r16 + lm;
#pragma unroll
        for (int r = 0; r < 8; ++r)
            ob[(r + (hi << 3)) * HW] = acc[r] + sh[r];
    }
}

extern "C" void kernel_launch(void* const* d_in, const int* in_sizes, int n_in,
                              void* d_out, int out_size, void* d_ws, size_t ws_size,
                              hipStream_t stream)
{
    const float* X    = (const float*)d_in[0];
    const float* wgt  = (const float*)d_in[1];
    const float* bia  = (const float*)d_in[2];
    float*       out  = (float*)d_out;
    float*       ws   = (float*)d_ws;   // uses ~1.13 MB

    dbn_pass1<<<dim3(GRP * BPG), dim3(256), 0, stream>>>(X, ws);
    dbn_pass2<<<dim3(GRP),       dim3(256), 0, stream>>>(ws, wgt, bia);
    dbn_pass3<<<dim3(GRP * BPG), dim3(256), 0, stream>>>(X, ws, out);
}